// ProposalModule_88081189307029
// MI455X (gfx1250) — compile-verified
//
#include <hip/hip_runtime.h>
#include <math.h>

// ---------------- problem constants (match reference) ----------------
#define BB 16
#define NN 16384
#define CC 256
#define PP 256
#define NSAMP 16
#define R2 0.09f               // RADIUS^2
#define PN2 128
#define MIDF 123
#define OUTF 119
#define K0W 259                // 3 + C
#define K0PAD 288              // padded to multiple of 32 (9 chunks)
#define BNEPS 1e-5f
#define M_MLP (BB*PP*NSAMP)    // 65536
#define M_FC  (BB*PP)          // 4096

typedef __attribute__((ext_vector_type(2))) float v2f;
typedef __attribute__((ext_vector_type(8))) float v8f;

// ---------------------------------------------------------------------
// 1) Furthest point sampling: one block per batch, 256 threads,
//    64 points per thread (interleaved), LDS argmax tree per iteration.
// ---------------------------------------------------------------------
__global__ __launch_bounds__(256)
void fps_kernel(const float* __restrict__ seed_pos, int* __restrict__ idx_out)
{
    const int b = blockIdx.x;
    const int tid = threadIdx.x;
    const float* sp = seed_pos + (size_t)b * NN * 3;

    __shared__ float s_far[3];
    __shared__ int   s_faridx;
    __shared__ float s_vals[256];
    __shared__ int   s_idx[256];

    float mind[NN / 256];
    #pragma unroll
    for (int j = 0; j < NN / 256; ++j) mind[j] = 1e10f;

    if (tid == 0) {
        s_faridx = 0;
        s_far[0] = sp[0]; s_far[1] = sp[1]; s_far[2] = sp[2];
    }
    __syncthreads();

    for (int it = 0; it < PP; ++it) {
        if (tid == 0) idx_out[(size_t)b * PP + it] = s_faridx;
        const float fx = s_far[0], fy = s_far[1], fz = s_far[2];

        float best = -1.0f; int bidx = 0;
        #pragma unroll 4
        for (int j = 0; j < NN / 256; ++j) {
            const int pt = j * 256 + tid;
            const float dx = sp[pt * 3 + 0] - fx;
            const float dy = sp[pt * 3 + 1] - fy;
            const float dz = sp[pt * 3 + 2] - fz;
            const float d  = dx * dx + dy * dy + dz * dz;
            mind[j] = fminf(mind[j], d);
            if (mind[j] > best) { best = mind[j]; bidx = pt; }
        }
        s_vals[tid] = best; s_idx[tid] = bidx;
        __syncthreads();
        for (int s = 128; s > 0; s >>= 1) {
            if (tid < s && s_vals[tid + s] > s_vals[tid]) {
                s_vals[tid] = s_vals[tid + s];
                s_idx[tid]  = s_idx[tid + s];
            }
            __syncthreads();
        }
        if (tid == 0) {
            const int f = s_idx[0];
            s_faridx = f;
            s_far[0] = sp[f * 3 + 0]; s_far[1] = sp[f * 3 + 1]; s_far[2] = sp[f * 3 + 2];
        }
        __syncthreads();
    }
}

// ---------------------------------------------------------------------
// 2) Ball query: one wave per center; ordered first-NS-in-radius via
//    wave32 ballot + prefix popcount; pad with first hit (or 0).
// ---------------------------------------------------------------------
__global__ __launch_bounds__(256)
void ballquery_kernel(const float* __restrict__ pos, const int* __restrict__ idx_p,
                      int* __restrict__ idx_out)
{
    const int wid  = (blockIdx.x * blockDim.x + threadIdx.x) >> 5;   // bp
    const int lane = threadIdx.x & 31;
    if (wid >= BB * PP) return;

    const int b = wid / PP;
    const float* pb = pos + (size_t)b * NN * 3;
    const int ci = idx_p[wid];
    const float cx = pb[ci * 3 + 0], cy = pb[ci * 3 + 1], cz = pb[ci * 3 + 2];

    int* out = idx_out + (size_t)wid * NSAMP;
    int cnt = 0, firsthit = 0;

    for (int c0 = 0; c0 < NN; c0 += 32) {
        const int pt = c0 + lane;
        const float dx = pb[pt * 3 + 0] - cx;
        const float dy = pb[pt * 3 + 1] - cy;
        const float dz = pb[pt * 3 + 2] - cz;
        const bool hit = (dx * dx + dy * dy + dz * dz) <= R2;
        const unsigned mm = (unsigned)__ballot(hit);
        const int before = __popc(mm & ((1u << lane) - 1u));
        if (hit && (cnt + before) < NSAMP) out[cnt + before] = pt;
        const int h = __popc(mm);
        if (cnt == 0 && h > 0) firsthit = c0 + __ffs(mm) - 1;
        cnt += h;
        if (cnt >= NSAMP) break;
    }
    if (cnt < NSAMP) {
        const int pad = (cnt > 0) ? firsthit : 0;
        for (int s = cnt + lane; s < NSAMP; s += 32) out[s] = pad;
    }
}

// ---------------------------------------------------------------------
// 3a) Feature transpose [B,C,N] -> [B,N,C] so gathers read 1KB rows.
// ---------------------------------------------------------------------
__global__ __launch_bounds__(256)
void transpose_kernel(const float* __restrict__ features, float* __restrict__ FT)
{
    __shared__ float tile[32][33];
    const int b  = blockIdx.z;
    const int n0 = blockIdx.x * 32;
    const int c0 = blockIdx.y * 32;
    const int tx = threadIdx.x, ty = threadIdx.y;   // 32 x 8

    const float* src = features + ((size_t)b * CC + c0) * NN + n0;
    #pragma unroll
    for (int j = 0; j < 32; j += 8)
        tile[ty + j][tx] = src[(size_t)(ty + j) * NN + tx];
    __syncthreads();
    float* dst = FT + ((size_t)b * NN + n0) * CC + c0;
    #pragma unroll
    for (int j = 0; j < 32; j += 8)
        dst[(size_t)(ty + j) * CC + tx] = tile[tx][ty + j];
}

// ---------------------------------------------------------------------
// 3b) Gather: build X0[M_MLP][K0PAD] = [gp - center | gathered feats | 0-pad]
// ---------------------------------------------------------------------
__global__ __launch_bounds__(256)
void gather_kernel(const float* __restrict__ pos, const float* __restrict__ FT,
                   const int* __restrict__ idx_p, const int* __restrict__ idx,
                   float* __restrict__ X0)
{
    const int bp = blockIdx.x;
    const int b  = bp / PP;
    const float* pb = pos + (size_t)b * NN * 3;
    const int ci = idx_p[bp];
    const float cx = pb[ci * 3 + 0], cy = pb[ci * 3 + 1], cz = pb[ci * 3 + 2];
    const float* ftb = FT + (size_t)b * NN * CC;

    for (int s = 0; s < NSAMP; ++s) {
        const int pt = idx[(size_t)bp * NSAMP + s];
        const float* frow = ftb + (size_t)pt * CC;
        float* row = X0 + ((size_t)bp * NSAMP + s) * K0PAD;
        for (int t = threadIdx.x; t < K0PAD; t += blockDim.x) {
            float v;
            if (t == 0)            v = pb[pt * 3 + 0] - cx;
            else if (t == 1)       v = pb[pt * 3 + 1] - cy;
            else if (t == 2)       v = pb[pt * 3 + 2] - cz;
            else if (t < 3 + CC)   v = frow[t - 3];
            else                   v = 0.0f;
            row[t] = v;
        }
    }
}

// ---------------------------------------------------------------------
// 4a) Weight prep: zero-padded Wp[Kceil][128] and bpad[128].
//     Moves ALL bounds checks out of the GEMM hot loop.
// ---------------------------------------------------------------------
__global__ __launch_bounds__(256)
void prep_weights_kernel(const float* __restrict__ W, const float* __restrict__ bias,
                         int Kw, int nout, int Kceil,
                         float* __restrict__ Wp, float* __restrict__ bpad)
{
    const int gid = blockIdx.x * 256 + threadIdx.x;
    if (gid < Kceil * 128) {
        const int k = gid >> 7, n = gid & 127;
        Wp[gid] = (k < Kw && n < nout) ? W[(size_t)k * nout + n] : 0.0f;
    }
    if (gid < 128) bpad[gid] = (gid < nout) ? bias[gid] : 0.0f;
}

// ---------------------------------------------------------------------
// 4b) fp32 WMMA GEMM + per-channel stats for global batchnorm.
//     Y[M,128] = act(X)[M,Kpad] @ Wp[Kpad][128] + bpad
//     act(x) = relu(x*act_a[k] + act_c[k]) on load if use_act.
//     Block = 256 threads = 8 waves; block owns 32 rows (2 M-tiles/wave),
//     wave w owns N-tile w. Branch-free inner loop: LDS -> 16x v_wmma.
// ---------------------------------------------------------------------
__global__ __launch_bounds__(256)
void gemm_bn_kernel(const float* __restrict__ X, int ldx, int Kpad,
                    const float* __restrict__ Wp, const float* __restrict__ bpad,
                    const float* __restrict__ act_a, const float* __restrict__ act_c,
                    int use_act,
                    float* __restrict__ Y,
                    float* __restrict__ sum, float* __restrict__ sumsq)
{
    __shared__ __align__(16) float As[32][36];     // 32 rows x KT(=32) cols (+pad)
    __shared__ __align__(16) float Bs[32][128];    // KT x 128 weight panel
    __shared__ float Aa[128], Ac[128];

    const int tid  = threadIdx.x;
    const int lane = tid & 31;
    const int half = lane >> 4;        // 0: K+{0,1}  1: K+{2,3}
    const int l16  = lane & 15;
    const int col  = ((tid >> 5) << 4) + l16;
    const int mbase = blockIdx.x * 32;

    if (use_act && tid < 128) { Aa[tid] = act_a[tid]; Ac[tid] = act_c[tid]; }

    const float* Xr = X + (size_t)mbase * ldx;
    v8f acc0 = {}, acc1 = {};

    const int ar = tid >> 3;           // 0..31  (A-tile row)
    const int ac = (tid & 7) * 4;      // 0..28  (A-tile col, float4)

    for (int k0 = 0; k0 < Kpad; k0 += 32) {
        __syncthreads();
        // ---- A tile: 32x32, one float4 per thread, act applied on load
        {
            float4 v = *(const float4*)(Xr + (size_t)ar * ldx + k0 + ac);
            if (use_act) {
                v.x = fmaxf(v.x * Aa[k0 + ac + 0] + Ac[k0 + ac + 0], 0.0f);
                v.y = fmaxf(v.y * Aa[k0 + ac + 1] + Ac[k0 + ac + 1], 0.0f);
                v.z = fmaxf(v.z * Aa[k0 + ac + 2] + Ac[k0 + ac + 2], 0.0f);
                v.w = fmaxf(v.w * Aa[k0 + ac + 3] + Ac[k0 + ac + 3], 0.0f);
            }
            As[ar][ac + 0] = v.x; As[ar][ac + 1] = v.y;
            As[ar][ac + 2] = v.z; As[ar][ac + 3] = v.w;
        }
        // ---- B panel: 32x128, four float4 per thread, coalesced
        {
            const float4* src = (const float4*)(Wp + (size_t)k0 * 128);
            float4* dst = (float4*)(&Bs[0][0]);
            #pragma unroll
            for (int i = 0; i < 4; ++i) dst[tid + i * 256] = src[tid + i * 256];
        }
        __syncthreads();

        #pragma unroll
        for (int kk = 0; kk < 32; kk += 4) {
            const int kr = kk + half * 2;
            v2f bf;
            bf.x = Bs[kr][col];
            bf.y = Bs[kr + 1][col];
            const v2f a0 = *(const v2f*)&As[l16][kr];
            const v2f a1 = *(const v2f*)&As[l16 + 16][kr];
            acc0 = __builtin_amdgcn_wmma_f32_16x16x4_f32(false, a0, false, bf, (short)0, acc0, false, false);
            acc1 = __builtin_amdgcn_wmma_f32_16x16x4_f32(false, a1, false, bf, (short)0, acc1, false, false);
        }
    }

    const float bval = bpad[col];
    float s = 0.0f, ss = 0.0f;
    float* Yt = Y + (size_t)mbase * 128;
    #pragma unroll
    for (int i = 0; i < 8; ++i) {
        const float v = acc0[i] + bval;     // padded cols: acc==0, bval==0
        Yt[(size_t)(i + half * 8) * 128 + col] = v;
        s += v; ss += v * v;
        const float w = acc1[i] + bval;
        Yt[(size_t)(16 + i + half * 8) * 128 + col] = w;
        s += w; ss += w * w;
    }
    s  += __shfl_xor(s, 16, 32);
    ss += __shfl_xor(ss, 16, 32);
    if (lane < 16) {
        atomicAdd(&sum[col],   s);
        atomicAdd(&sumsq[col], ss);
    }
}

// ---------------------------------------------------------------------
// 5) BN stat helpers
// ---------------------------------------------------------------------
__global__ void zero_stats_kernel(float* sum, float* sumsq)
{
    const int t = threadIdx.x;                  // 128
    sum[t] = 0.0f; sumsq[t] = 0.0f;
}

__global__ void finalize_stats_kernel(const float* __restrict__ sum,
                                      const float* __restrict__ sumsq,
                                      const float* __restrict__ g,
                                      const float* __restrict__ be,
                                      float* __restrict__ act_a,
                                      float* __restrict__ act_c,
                                      int nout, float invM)
{
    const int t = threadIdx.x;                  // 128
    if (t < nout) {
        const float m = sum[t] * invM;
        const float v = sumsq[t] * invM - m * m;
        const float s = g[t] * rsqrtf(v + BNEPS);
        act_a[t] = s;
        act_c[t] = be[t] - m * s;
    } else {
        act_a[t] = 0.0f; act_c[t] = 0.0f;
    }
}

// ---------------------------------------------------------------------
// 6) Max over NS neighbors (applies act+relu of last MLP layer).
// ---------------------------------------------------------------------
__global__ __launch_bounds__(128)
void maxpool_kernel(const float* __restrict__ Y, const float* __restrict__ act_a,
                    const float* __restrict__ act_c, float* __restrict__ F)
{
    const int bp = blockIdx.x;
    const int ch = threadIdx.x;                 // 128
    const float a = act_a[ch], c = act_c[ch];
    float m = 0.0f;                             // relu outputs are >= 0
    #pragma unroll
    for (int s = 0; s < NSAMP; ++s) {
        const float v = fmaxf(Y[((size_t)bp * NSAMP + s) * 128 + ch] * a + c, 0.0f);
        m = fmaxf(m, v);
    }
    F[(size_t)bp * 128 + ch] = m;
}

// ---------------------------------------------------------------------
// 7) Final conv (123->119) + bias + transposed store [B, OUTF, P].
// ---------------------------------------------------------------------
__global__ __launch_bounds__(128)
void c3_kernel(const float* __restrict__ X, const float* __restrict__ act_a,
               const float* __restrict__ act_c, const float* __restrict__ W,
               const float* __restrict__ bias, float* __restrict__ out)
{
    const int bp = blockIdx.x;
    const int b  = bp / PP, p = bp % PP;
    __shared__ float xr[128];
    for (int t = threadIdx.x; t < 128; t += blockDim.x)
        xr[t] = fmaxf(X[(size_t)bp * 128 + t] * act_a[t] + act_c[t], 0.0f);
    __syncthreads();
    for (int o = threadIdx.x; o < OUTF; o += blockDim.x) {
        float s = bias[o];
        for (int k = 0; k < MIDF; ++k)
            s += xr[k] * W[(size_t)k * OUTF + o];
        out[((size_t)b * OUTF + o) * PP + p] = s;
    }
}

// ---------------------------------------------------------------------
// host-side launcher
// ---------------------------------------------------------------------
extern "C" void kernel_launch(void* const* d_in, const int* in_sizes, int n_in,
                              void* d_out, int out_size, void* d_ws, size_t ws_size,
                              hipStream_t stream)
{
    (void)in_sizes; (void)n_in; (void)out_size; (void)ws_size;

    const float* pos      = (const float*)d_in[0];
    const float* features = (const float*)d_in[1];
    const float* seed_pos = (const float*)d_in[2];
    const float* mlp_w[3] = { (const float*)d_in[3],  (const float*)d_in[7],  (const float*)d_in[11] };
    const float* mlp_b[3] = { (const float*)d_in[4],  (const float*)d_in[8],  (const float*)d_in[12] };
    const float* mlp_g[3] = { (const float*)d_in[5],  (const float*)d_in[9],  (const float*)d_in[13] };
    const float* mlp_be[3]= { (const float*)d_in[6],  (const float*)d_in[10], (const float*)d_in[14] };
    const float* c1_w = (const float*)d_in[15]; const float* c1_b = (const float*)d_in[16];
    const float* bn1_g= (const float*)d_in[17]; const float* bn1_b= (const float*)d_in[18];
    const float* c2_w = (const float*)d_in[19]; const float* c2_b = (const float*)d_in[20];
    const float* bn2_g= (const float*)d_in[21]; const float* bn2_b= (const float*)d_in[22];
    const float* c3_w = (const float*)d_in[23]; const float* c3_b = (const float*)d_in[24];
    float* out = (float*)d_out;

    // workspace carve-up
    char* ws = (char*)d_ws;
    size_t off = 0;
    auto carve = [&](size_t bytes) { void* p = ws + off; off += (bytes + 255) & ~(size_t)255; return p; };
    int*   idx_p = (int*)  carve((size_t)BB * PP * 4);
    int*   idxBQ = (int*)  carve((size_t)BB * PP * NSAMP * 4);
    float* FT    = (float*)carve((size_t)BB * NN * CC * 4);
    float* X0    = (float*)carve((size_t)M_MLP * K0PAD * 4);
    float* Y0    = (float*)carve((size_t)M_MLP * 128 * 4);
    float* Y1    = (float*)carve((size_t)M_MLP * 128 * 4);
    float* F     = (float*)carve((size_t)M_FC * 128 * 4);
    float* G1    = (float*)carve((size_t)M_FC * 128 * 4);
    float* G2    = (float*)carve((size_t)M_FC * 128 * 4);
    float* Wp    = (float*)carve((size_t)K0PAD * 128 * 4);
    float* bpad  = (float*)carve(128 * 4);
    float* ssum  = (float*)carve(128 * 4);
    float* ssq   = (float*)carve(128 * 4);
    float* act_a = (float*)carve(128 * 4);
    float* act_c = (float*)carve(128 * 4);

    // 1) FPS
    fps_kernel<<<BB, 256, 0, stream>>>(seed_pos, idx_p);
    // 2) Ball query (8 waves/block)
    ballquery_kernel<<<(BB * PP) / 8, 256, 0, stream>>>(pos, idx_p, idxBQ);
    // 3) transpose features + gather
    transpose_kernel<<<dim3(NN / 32, CC / 32, BB), dim3(32, 8), 0, stream>>>(features, FT);
    gather_kernel<<<BB * PP, 256, 0, stream>>>(pos, FT, idx_p, idxBQ, X0);

    const float invM_mlp = 1.0f / (float)M_MLP;
    const float invM_fc  = 1.0f / (float)M_FC;
    const int prep_full = (K0PAD * 128 + 255) / 256;
    const int prep_128  = (128 * 128 + 255) / 256;

    // 4) shared MLP: 3 x (prep + GEMM + global BN), ping-pong Y0/Y1
    prep_weights_kernel<<<prep_full, 256, 0, stream>>>(mlp_w[0], mlp_b[0], K0W, PN2, K0PAD, Wp, bpad);
    zero_stats_kernel<<<1, 128, 0, stream>>>(ssum, ssq);
    gemm_bn_kernel<<<M_MLP / 32, 256, 0, stream>>>(X0, K0PAD, K0PAD, Wp, bpad,
                                                   act_a, act_c, 0, Y0, ssum, ssq);
    finalize_stats_kernel<<<1, 128, 0, stream>>>(ssum, ssq, mlp_g[0], mlp_be[0], act_a, act_c, PN2, invM_mlp);

    prep_weights_kernel<<<prep_128, 256, 0, stream>>>(mlp_w[1], mlp_b[1], 128, PN2, 128, Wp, bpad);
    zero_stats_kernel<<<1, 128, 0, stream>>>(ssum, ssq);
    gemm_bn_kernel<<<M_MLP / 32, 256, 0, stream>>>(Y0, 128, 128, Wp, bpad,
                                                   act_a, act_c, 1, Y1, ssum, ssq);
    finalize_stats_kernel<<<1, 128, 0, stream>>>(ssum, ssq, mlp_g[1], mlp_be[1], act_a, act_c, PN2, invM_mlp);

    prep_weights_kernel<<<prep_128, 256, 0, stream>>>(mlp_w[2], mlp_b[2], 128, PN2, 128, Wp, bpad);
    zero_stats_kernel<<<1, 128, 0, stream>>>(ssum, ssq);
    gemm_bn_kernel<<<M_MLP / 32, 256, 0, stream>>>(Y1, 128, 128, Wp, bpad,
                                                   act_a, act_c, 1, Y0, ssum, ssq);
    finalize_stats_kernel<<<1, 128, 0, stream>>>(ssum, ssq, mlp_g[2], mlp_be[2], act_a, act_c, PN2, invM_mlp);

    // 5) max over neighbors (applies layer-2 act+relu)
    maxpool_kernel<<<M_FC, 128, 0, stream>>>(Y0, act_a, act_c, F);

    // 6) FC head: c1 (128->128) and c2 (128->123), each with global BN
    prep_weights_kernel<<<prep_128, 256, 0, stream>>>(c1_w, c1_b, 128, PN2, 128, Wp, bpad);
    zero_stats_kernel<<<1, 128, 0, stream>>>(ssum, ssq);
    gemm_bn_kernel<<<M_FC / 32, 256, 0, stream>>>(F, 128, 128, Wp, bpad,
                                                  act_a, act_c, 0, G1, ssum, ssq);
    finalize_stats_kernel<<<1, 128, 0, stream>>>(ssum, ssq, bn1_g, bn1_b, act_a, act_c, PN2, invM_fc);

    prep_weights_kernel<<<prep_128, 256, 0, stream>>>(c2_w, c2_b, 128, MIDF, 128, Wp, bpad);
    zero_stats_kernel<<<1, 128, 0, stream>>>(ssum, ssq);
    gemm_bn_kernel<<<M_FC / 32, 256, 0, stream>>>(G1, 128, 128, Wp, bpad,
                                                  act_a, act_c, 1, G2, ssum, ssq);
    finalize_stats_kernel<<<1, 128, 0, stream>>>(ssum, ssq, bn2_g, bn2_b, act_a, act_c, MIDF, invM_fc);

    // 7) final conv + transpose to [B, OUTF, P]
    c3_kernel<<<M_FC, 128, 0, stream>>>(G2, act_a, act_c, c3_w, c3_b, out);
}